// PMF_455266533951
// MI455X (gfx1250) — compile-verified
//
#include <hip/hip_runtime.h>
#include <hip/hip_bf16.h>

typedef __attribute__((ext_vector_type(2))) float v2f;
typedef __attribute__((ext_vector_type(8))) float v8f;

#define PMF_DIM 64
#define PMF_LAMBDA 0.01f

__global__ void pmf_zero_kernel(float* out) { out[0] = 0.0f; }

// One wave (32 lanes) processes a tile of 16 batch rows.
// D = ue_tile (16x64) @ ie_tile^T (64x16) accumulated via 16 steps of
// v_wmma_f32_16x16x4_f32 (full f32 precision). diag(D) = per-row dots.
__global__ __launch_bounds__(256) void pmf_loss_kernel(
    const int* __restrict__ u_idx,
    const int* __restrict__ i_idx,
    const int* __restrict__ s_val,
    const float* __restrict__ W,
    const float* __restrict__ H,
    float* __restrict__ out,
    int batch)
{
    const int lane   = threadIdx.x & 31;
    const int wave   = (blockIdx.x * blockDim.x + threadIdx.x) >> 5;
    const int nwaves = (gridDim.x * blockDim.x) >> 5;
    const int ntiles = batch >> 4;           // 16 rows per tile
    const int r      = lane & 15;            // M (for A) / N (for B) index
    const int h      = lane >> 4;            // lane half selects K sub-pair
    const float inv_batch = 1.0f / (float)batch;

    float acc = 0.0f;

    for (int tile = wave; tile < ntiles; tile += nwaves) {
        const int base = tile << 4;
        const int ui = u_idx[base + r];
        const int ii = i_idx[base + r];
        const float* __restrict__ Wr = W + (size_t)ui * PMF_DIM;
        const float* __restrict__ Hr = H + (size_t)ii * PMF_DIM;

        v8f c = {0.f, 0.f, 0.f, 0.f, 0.f, 0.f, 0.f, 0.f};
        float reg = 0.0f;

        // k-loop: A layout VGPR0/1 = K = 4t+2h, 4t+2h+1 (lanes 0-15: h=0 -> K=0,1;
        // lanes 16-31: h=1 -> K=2,3), identical pattern for B. Contiguous float2 loads.
#pragma unroll
        for (int t = 0; t < PMF_DIM / 4; ++t) {
            const int k = 4 * t + 2 * h;
            const v2f a = *(const v2f*)(Wr + k);   // 8B aligned (k even, rows 256B)
            const v2f b = *(const v2f*)(Hr + k);
            // every tile element touched exactly once across lanes/steps
            reg = fmaf(a.x, a.x, fmaf(a.y, a.y, fmaf(b.x, b.x, fmaf(b.y, b.y, reg))));
            // (neg_a, A, neg_b, B, c_mod, C, reuse_a, reuse_b)
            c = __builtin_amdgcn_wmma_f32_16x16x4_f32(
                    false, a, false, b, (short)0, c, false, false);
        }

        // diag(D): row m in [0,8): lane m, c[m]; row m in [8,16): lane m+16, c[m-8]
        const int sel = (lane < 8) ? lane : (lane - 24);
        float x = c[0];
        x = (sel == 1) ? c[1] : x;
        x = (sel == 2) ? c[2] : x;
        x = (sel == 3) ? c[3] : x;
        x = (sel == 4) ? c[4] : x;
        x = (sel == 5) ? c[5] : x;
        x = (sel == 6) ? c[6] : x;
        x = (sel == 7) ? c[7] : x;

        float lp = 0.0f;
        if (lane < 8 || lane >= 24) {
            const int row = (lane < 8) ? lane : (lane - 16);
            const float sv = (float)s_val[base + row];
            // stable log_sigmoid(x) = min(x,0) - log1p(exp(-|x|))
            const float ls = fminf(x, 0.0f) - log1pf(expf(-fabsf(x)));
            const float d = sv - ls;
            lp = d * d;
        }
        acc += lp + PMF_LAMBDA * reg;
    }

    // wave32 butterfly reduction
#pragma unroll
    for (int off = 16; off > 0; off >>= 1)
        acc += __shfl_xor(acc, off, 32);

    if (lane == 0)
        atomicAdd(out, acc * inv_batch);
}

extern "C" void kernel_launch(void* const* d_in, const int* in_sizes, int n_in,
                              void* d_out, int out_size, void* d_ws, size_t ws_size,
                              hipStream_t stream) {
    const int*   u = (const int*)d_in[0];
    const int*   i = (const int*)d_in[1];
    const int*   s = (const int*)d_in[2];
    const float* W = (const float*)d_in[3];
    const float* H = (const float*)d_in[4];
    float* out = (float*)d_out;

    const int batch  = in_sizes[0];          // 16384
    const int ntiles = batch >> 4;           // 1024 tiles of 16 rows
    int blocks = (ntiles + 7) / 8;           // 8 waves per 256-thread block
    if (blocks > 128) blocks = 128;
    if (blocks < 1)   blocks = 1;

    pmf_zero_kernel<<<1, 1, 0, stream>>>(out);
    pmf_loss_kernel<<<blocks, 256, 0, stream>>>(u, i, s, W, H, out, batch);
}